// ComplexKLLoss_1597727834143
// MI455X (gfx1250) — compile-verified
//
#include <hip/hip_runtime.h>
#include <math.h>

#define BATCH   64
#define PLANE   (512 * 512)        // 262144 elements per plane
#define BINS    100
#define CHUNKS  64                 // blocks per batch for the streaming passes
#define BLOCK   256                // 8 wave32s
#define EPSF    1e-10f
#define PI_F    3.14159265358979323846f
#define LN2_F   0.69314718056f

typedef __attribute__((ext_vector_type(2))) float v2f;
typedef __attribute__((ext_vector_type(8))) float v8f;

// ---------------------------------------------------------------------------
// 32-lane wave sum via V_WMMA_F32_16X16X4_F32.
// A (16x4 layout): row m of A holds {x[m], 0, x[m+16], 0}; B = ones =>
// D[m,n] = x[m] + x[m+16]. Per-lane sum of the 8 C/D VGPRs leaves half-sums
// in lanes 0-15 / 16-31; one xor-16 shuffle finishes the 32-lane reduction.
// Requires EXEC == all ones (call only from fully-active code).
// ---------------------------------------------------------------------------
__device__ __forceinline__ float wave_sum_wmma(float x) {
  v2f a; a.x = x;    a.y = 0.0f;
  v2f b; b.x = 1.0f; b.y = 1.0f;
  v8f c = {};
  c = __builtin_amdgcn_wmma_f32_16x16x4_f32(false, a, false, b, (short)0, c,
                                            false, false);
  float s = c[0] + c[1] + c[2] + c[3] + c[4] + c[5] + c[6] + c[7];
  s += __shfl_xor(s, 16, 32);
  return s;
}

// Deterministic block-wide sum: WMMA wave reduce + fixed LDS tree.
__device__ __forceinline__ float block_sum(float x, float* slots, int nwaves) {
  float w = wave_sum_wmma(x);
  int lane = threadIdx.x & 31;
  int wid  = threadIdx.x >> 5;
  __syncthreads();                 // protect slots reuse across calls
  if (lane == 0) slots[wid] = w;
  __syncthreads();
  float tot = 0.0f;
  for (int i = 0; i < nwaves; ++i) tot += slots[i];
  return tot;                      // broadcast to all threads
}

// Single-instruction transcendental helpers (v_rcp/v_sqrt/v_log).
__device__ __forceinline__ float fast_rcp(float x)  { return __builtin_amdgcn_rcpf(x); }
__device__ __forceinline__ float fast_sqrt(float x) { return __builtin_amdgcn_sqrtf(x); }
__device__ __forceinline__ float fast_ln(float x)   { return __builtin_amdgcn_logf(x) * LN2_F; }
__device__ __forceinline__ float fast_log2(float x) { return __builtin_amdgcn_logf(x); }

// Fast atan2: only needs ~pi/100 accuracy for 100-bin histogramming.
// Division replaced by v_rcp; copysign done with bit ops (no v_cmp_class).
__device__ __forceinline__ float fast_atan2f(float y, float x) {
  float ax = fabsf(x), ay = fabsf(y);
  float mx = fmaxf(ax, ay), mn = fminf(ax, ay);
  float a  = mn * fast_rcp(mx);
  a = (mx > 0.0f) ? a : 0.0f;      // guard 0/0 -> angle 0
  float s  = a * a;
  float r  = a * fmaf(s, fmaf(s, fmaf(s, fmaf(s, fmaf(s, -0.0117212f,
                    0.05265332f), -0.11643287f), 0.19354346f),
                    -0.33262347f), 0.99997726f);
  if (ay > ax)   r = 1.57079632679f - r;
  if (x < 0.0f)  r = 3.14159265359f - r;
  unsigned ur = (__float_as_uint(r) & 0x7fffffffu) |
                (__float_as_uint(y) & 0x80000000u);
  return __uint_as_float(ur);
}

__device__ __forceinline__ int phase_bin(float im, float re) {
  float ph = fast_atan2f(im, re);
  float t  = (ph + PI_F) * 15.9154943092f;   // * BINS / (2*pi)
  int   bi = (int)t;
  bi = bi < 0 ? 0 : bi;
  return bi > (BINS - 1) ? (BINS - 1) : bi;
}

// ---------------------------------------------------------------------------
// Pass 1: stream A,B once; per batch accumulate sum(|A|), sum(|B|) partials
// and 100-bin phase histograms (LDS u32 atomics -> global u32 atomics).
// ---------------------------------------------------------------------------
__global__ void pass1_kernel(const float* __restrict__ A,
                             const float* __restrict__ B,
                             float* __restrict__ sumPartA,
                             float* __restrict__ sumPartB,
                             unsigned* __restrict__ gHistA,
                             unsigned* __restrict__ gHistB) {
  __shared__ unsigned hA[BINS], hB[BINS];
  __shared__ float    slots[BLOCK / 32];
  const int t     = threadIdx.x;
  const int chunk = blockIdx.x;
  const int b     = blockIdx.y;
  if (t < BINS) { hA[t] = 0u; hB[t] = 0u; }
  __syncthreads();

  const float* Are = A + (size_t)b * 2 * PLANE;
  const float* Aim = Are + PLANE;
  const float* Bre = B + (size_t)b * 2 * PLANE;
  const float* Bim = Bre + PLANE;

  float sA = 0.0f, sB = 0.0f;
  const int base = chunk * (PLANE / CHUNKS) + t * 4;   // 4096 elems per chunk
#pragma unroll
  for (int i = 0; i < 4; ++i) {
    const int idx = base + i * (BLOCK * 4);
    if (i < 3) {                                       // global_prefetch path
      __builtin_prefetch(Are + idx + BLOCK * 4, 0, 1);
      __builtin_prefetch(Bre + idx + BLOCK * 4, 0, 1);
    }
    float4 ar = *(const float4*)(Are + idx);
    float4 ai = *(const float4*)(Aim + idx);
    float4 br = *(const float4*)(Bre + idx);
    float4 bi = *(const float4*)(Bim + idx);
    float arv[4] = {ar.x, ar.y, ar.z, ar.w};
    float aiv[4] = {ai.x, ai.y, ai.z, ai.w};
    float brv[4] = {br.x, br.y, br.z, br.w};
    float biv[4] = {bi.x, bi.y, bi.z, bi.w};
#pragma unroll
    for (int k = 0; k < 4; ++k) {
      sA += fast_sqrt(fmaf(arv[k], arv[k], aiv[k] * aiv[k]));
      sB += fast_sqrt(fmaf(brv[k], brv[k], biv[k] * biv[k]));
      atomicAdd(&hA[phase_bin(aiv[k], arv[k])], 1u);
      atomicAdd(&hB[phase_bin(biv[k], brv[k])], 1u);
    }
  }

  float totA = block_sum(sA, slots, BLOCK / 32);
  float totB = block_sum(sB, slots, BLOCK / 32);
  if (t == 0) {
    sumPartA[b * CHUNKS + chunk] = totA;
    sumPartB[b * CHUNKS + chunk] = totB;
  }
  __syncthreads();
  if (t < BINS) {
    atomicAdd(&gHistA[b * BINS + t], hA[t]);   // u32: order-independent
    atomicAdd(&gHistB[b * BINS + t], hB[t]);
  }
}

// Reduce 64 per-chunk partials -> per-batch amplitude sums (deterministic tree).
__global__ void reduce_sums_kernel(const float* __restrict__ sumPartA,
                                   const float* __restrict__ sumPartB,
                                   float* __restrict__ sumA,
                                   float* __restrict__ sumB) {
  __shared__ float slots[2];
  const int b = blockIdx.x, t = threadIdx.x;   // 64 threads = 2 full waves
  float pa = sumPartA[b * CHUNKS + t];
  float pb = sumPartB[b * CHUNKS + t];
  float ta = block_sum(pa, slots, 2);
  float tb = block_sum(pb, slots, 2);
  if (t == 0) { sumA[b] = ta; sumB[b] = tb; }
}

// ---------------------------------------------------------------------------
// Pass 2: stream A,B again (reverse batch order to catch L2-resident tail of
// pass 1); per element P*(ln(P+e)-ln(Q+e)) via two v_log_f32 — no division.
// ---------------------------------------------------------------------------
__global__ void pass2_kernel(const float* __restrict__ A,
                             const float* __restrict__ B,
                             const float* __restrict__ sumA,
                             const float* __restrict__ sumB,
                             float* __restrict__ klPart) {
  __shared__ float slots[BLOCK / 32];
  const int t     = threadIdx.x;
  const int chunk = blockIdx.x;
  const int b     = (BATCH - 1) - blockIdx.y;  // reverse: L2 reuse from pass 1

  const float invSA = 1.0f / (sumA[b] + EPSF);
  const float invSB = 1.0f / (sumB[b] + EPSF);

  const float* Are = A + (size_t)b * 2 * PLANE;
  const float* Aim = Are + PLANE;
  const float* Bre = B + (size_t)b * 2 * PLANE;
  const float* Bim = Bre + PLANE;

  float kl = 0.0f;
  const int base = chunk * (PLANE / CHUNKS) + t * 4;
#pragma unroll
  for (int i = 0; i < 4; ++i) {
    const int idx = base + i * (BLOCK * 4);
    if (i < 3) {
      __builtin_prefetch(Aim + idx + BLOCK * 4, 0, 1);
      __builtin_prefetch(Bim + idx + BLOCK * 4, 0, 1);
    }
    float4 ar = *(const float4*)(Are + idx);
    float4 ai = *(const float4*)(Aim + idx);
    float4 br = *(const float4*)(Bre + idx);
    float4 bi = *(const float4*)(Bim + idx);
    float arv[4] = {ar.x, ar.y, ar.z, ar.w};
    float aiv[4] = {ai.x, ai.y, ai.z, ai.w};
    float brv[4] = {br.x, br.y, br.z, br.w};
    float biv[4] = {bi.x, bi.y, bi.z, bi.w};
#pragma unroll
    for (int k = 0; k < 4; ++k) {
      float P = fast_sqrt(fmaf(arv[k], arv[k], aiv[k] * aiv[k])) * invSA;
      float Q = fast_sqrt(fmaf(brv[k], brv[k], biv[k] * biv[k])) * invSB;
      // P * ln((P+e)/(Q+e)) = P * ln2 * (log2(P+e) - log2(Q+e))
      kl += P * (LN2_F * (fast_log2(P + EPSF) - fast_log2(Q + EPSF)));
    }
  }

  float tot = block_sum(kl, slots, BLOCK / 32);
  if (t == 0) klPart[b * CHUNKS + chunk] = tot;
}

// ---------------------------------------------------------------------------
// Final: reduce amp partials, compute phase-histogram KL (fp32 linspace edge
// widths, density -> renormalize -> KL summed over batch+bins), emit output.
// ---------------------------------------------------------------------------
__global__ void final_kernel(const float* __restrict__ klPart,
                             const unsigned* __restrict__ gHistA,
                             const unsigned* __restrict__ gHistB,
                             float* __restrict__ out) {
  __shared__ float slots[4];
  __shared__ float ampL[BATCH];
  __shared__ float phaseAcc;
  const int t = threadIdx.x;                   // 128 threads = 4 full waves

  if (t < BATCH) {
    float s = 0.0f;
    for (int c = 0; c < CHUNKS; ++c) s += klPart[t * CHUNKS + c];
    ampL[t] = s;
  }
  if (t == 0) phaseAcc = 0.0f;
  __syncthreads();

  const bool act = (t < BINS);
  float invW = 0.0f;
  if (act) {                                   // fp32 linspace(-pi, pi, 101) widths
    const double step = 6.283185307179586476925286766559 / 100.0;
    float e0 = (float)(-3.14159265358979323846 + step * (double)t);
    float e1 = (float)(-3.14159265358979323846 + step * (double)(t + 1));
    invW = 1.0f / (e1 - e0);                   // tiny kernel: accuracy over speed
  }
  const float nInv = 1.0f / (float)PLANE;

  for (int b = 0; b < BATCH; ++b) {
    float dA = 0.0f, dB = 0.0f;
    if (act) {
      dA = (float)gHistA[b * BINS + t] * nInv * invW;   // density
      dB = (float)gHistB[b * BINS + t] * nInv * invW;
    }
    float sdA = block_sum(dA, slots, 4);
    float sdB = block_sum(dB, slots, 4);
    float term = 0.0f;
    if (act) {
      float Pp = dA / (sdA + EPSF);
      float Qp = dB / (sdB + EPSF);
      term = Pp * logf((Pp + EPSF) / (Qp + EPSF));
    }
    float tsum = block_sum(term, slots, 4);
    if (t == 0) phaseAcc += tsum;
  }
  __syncthreads();
  if (t < BATCH) out[t] = 0.5f * ampL[t] + 0.5f * phaseAcc;
}

// ---------------------------------------------------------------------------
extern "C" void kernel_launch(void* const* d_in, const int* in_sizes, int n_in,
                              void* d_out, int out_size, void* d_ws,
                              size_t ws_size, hipStream_t stream) {
  (void)in_sizes; (void)n_in; (void)out_size; (void)ws_size;
  const float* A = (const float*)d_in[0];
  const float* B = (const float*)d_in[1];
  float* out = (float*)d_out;

  char* ws = (char*)d_ws;
  float*    sumPartA = (float*)ws;                       // 64*64
  float*    sumPartB = sumPartA + BATCH * CHUNKS;        // 64*64
  float*    klPart   = sumPartB + BATCH * CHUNKS;        // 64*64
  float*    sumA     = klPart   + BATCH * CHUNKS;        // 64
  float*    sumB     = sumA + BATCH;                     // 64
  unsigned* histA    = (unsigned*)(sumB + BATCH);        // 64*100
  unsigned* histB    = histA + BATCH * BINS;             // 64*100
  const size_t wsBytes = (size_t)((char*)(histB + BATCH * BINS) - ws);

  hipMemsetAsync(d_ws, 0, wsBytes, stream);              // graph-capture safe

  dim3 grid(CHUNKS, BATCH);
  pass1_kernel<<<grid, BLOCK, 0, stream>>>(A, B, sumPartA, sumPartB, histA, histB);
  reduce_sums_kernel<<<BATCH, 64, 0, stream>>>(sumPartA, sumPartB, sumA, sumB);
  pass2_kernel<<<grid, BLOCK, 0, stream>>>(A, B, sumA, sumB, klPart);
  final_kernel<<<1, 128, 0, stream>>>(klPart, histA, histB, out);
}